// CombinedNN_27350351741741
// MI455X (gfx1250) — compile-verified
//
#include <hip/hip_runtime.h>

#define N_NODES 100000
#define N_EDGES 800000
#define F_INF   75
#define HID     100
#define OUTF    3
#define BN_EPS  1e-5f

typedef __attribute__((ext_vector_type(2))) float v2f;
typedef __attribute__((ext_vector_type(8))) float v8f;

// ---------------------------------------------------------------- utilities
__global__ void fill_f32(float* __restrict__ p, int n, float v) {
  int i = blockIdx.x * blockDim.x + threadIdx.x;
  if (i < n) p[i] = v;
}

// deg[dst] += 1 per edge (deg pre-filled with 1.0 for the self-loop)
__global__ void degree_kernel(const int* __restrict__ dst, int e,
                              float* __restrict__ deg) {
  int i = blockIdx.x * blockDim.x + threadIdx.x;
  if (i < e) atomicAdd(&deg[dst[i]], 1.0f);
}

// in-place deg -> 1/sqrt(max(deg,1))
__global__ void rsqrt_kernel(float* __restrict__ d, int n) {
  int i = blockIdx.x * blockDim.x + threadIdx.x;
  if (i < n) d[i] = rsqrtf(fmaxf(d[i], 1.0f));
}

// ---------------------------------------------------------------- WMMA GEMM
// C[M x NCV] = A[M x KV] @ W[KV x NCV]   (row-major, ldA == KV, ldC == NCV)
// M must be a multiple of 16 (N_NODES = 6250 * 16).
// EPIL: 0 = plain store, 1 = relu(BN(y + bias)), 2 = relu(y + bias)
template <int KV, int NCV, int EPIL>
__global__ void gemm_wmma(const float* __restrict__ A,
                          const float* __restrict__ W,
                          float* __restrict__ C,
                          const float* __restrict__ bias,
                          const float* __restrict__ bn_g,
                          const float* __restrict__ bn_b,
                          const float* __restrict__ bn_m,
                          const float* __restrict__ bn_v) {
  constexpr int KP  = (KV + 3) & ~3;     // K padded to WMMA K-step
  constexpr int NT  = (NCV + 15) / 16;   // 16-col tiles
  constexpr int NTP = NT * 16;           // padded column count

  // W stored K-pair interleaved: [(k>>1)][n][k&1] -> B fragment = 1 b64 load
  __shared__ float ldsW[(KP / 2) * NTP * 2];
  // A tile row-major: [r][k] -> A fragment = 1 b64 load
  __shared__ float ldsA[16 * KP];

  const int mBase = blockIdx.x * 16;

  for (int idx = threadIdx.x; idx < KP * NTP; idx += blockDim.x) {
    const int k = idx / NTP;
    const int n = idx - k * NTP;
    const float w = (k < KV && n < NCV) ? W[k * NCV + n] : 0.0f;
    ldsW[(k >> 1) * (NTP * 2) + n * 2 + (k & 1)] = w;
  }
  for (int idx = threadIdx.x; idx < 16 * KP; idx += blockDim.x) {
    const int r = idx / KP;
    const int k = idx - r * KP;
    ldsA[idx] = (k < KV) ? A[(long)(mBase + r) * KV + k] : 0.0f;
  }
  __syncthreads();

  const int lane  = threadIdx.x & 31;
  const int wave  = threadIdx.x >> 5;
  const int nWav  = blockDim.x >> 5;
  const int kHalf = (lane >> 4) << 1;       // lanes 16-31 hold K+2,K+3
  const int nFrag = lane & 15;              // B/D column within tile
  const int mRowD = (lane >> 4) << 3;       // D rows: lanes 16-31 -> M+8..15
  const int aRow  = (lane & 15) * KP;       // this lane's A row base

  const float2* __restrict__ ldsA2 = reinterpret_cast<const float2*>(ldsA);
  const float2* __restrict__ ldsW2 = reinterpret_cast<const float2*>(ldsW);

  for (int t = wave; t < NT; t += nWav) {
    v8f acc = {0.f, 0.f, 0.f, 0.f, 0.f, 0.f, 0.f, 0.f};
    const int nCol = t * 16 + nFrag;

#pragma unroll
    for (int k = 0; k < KP; k += 4) {
      const int ka = k + kHalf;
      const float2 av = ldsA2[(aRow + ka) >> 1];        // A[m][ka], A[m][ka+1]
      const float2 bv = ldsW2[(ka >> 1) * NTP + nCol];  // W[ka][n], W[ka+1][n]
      v2f a; a.x = av.x; a.y = av.y;
      v2f b; b.x = bv.x; b.y = bv.y;
      acc = __builtin_amdgcn_wmma_f32_16x16x4_f32(
          /*neg_a=*/false, a, /*neg_b=*/false, b,
          /*c_mod=*/(short)0, acc, /*reuse_a=*/false, /*reuse_b=*/false);
    }

    if (nCol < NCV) {
      float bv = 0.f, gg = 1.f, bb = 0.f, mm = 0.f, rs = 1.f;
      if constexpr (EPIL >= 1) bv = bias[nCol];
      if constexpr (EPIL == 1) {
        gg = bn_g[nCol]; bb = bn_b[nCol]; mm = bn_m[nCol];
        rs = rsqrtf(bn_v[nCol] + BN_EPS);
      }
#pragma unroll
      for (int r = 0; r < 8; ++r) {
        const int row = mBase + mRowD + r;
        float y = acc[r];
        if constexpr (EPIL == 1) {        // relu(BN(y + b))
          y += bv;
          y = gg * (y - mm) * rs + bb;
          y = fmaxf(y, 0.f);
        } else if constexpr (EPIL == 2) { // relu(y + b)
          y = fmaxf(y + bv, 0.f);
        }
        C[(long)row * NCV + nCol] = y;
      }
    }
  }
}

// ---------------------------------------------------------------- edge scatter
// One wave per edge: agg[dst,:] += h[src,:] * dinv[src]*dinv[dst]
__global__ void edge_agg(const int* __restrict__ src, const int* __restrict__ dst,
                         int e, const float* __restrict__ dinv,
                         const float* __restrict__ h, float* __restrict__ agg) {
  const int wid  = (blockIdx.x * blockDim.x + threadIdx.x) >> 5;
  const int lane = threadIdx.x & 31;
  if (wid >= e) return;
  const int   s   = src[wid];
  const int   d   = dst[wid];
  const float nm  = dinv[s] * dinv[d];
  const long  so  = (long)s * HID;
  const long  dof = (long)d * HID;
  for (int f = lane; f < HID; f += 32)
    atomicAdd(&agg[dof + f], h[so + f] * nm);
}

// ---------------------------------------------------------------- GCN post
// in-place: agg = BN(relu(agg + h*dinv^2 (self-loop) + bias))
__global__ void gcn_post(float* __restrict__ agg, const float* __restrict__ h,
                         const float* __restrict__ dinv,
                         const float* __restrict__ b, const float* __restrict__ g,
                         const float* __restrict__ be, const float* __restrict__ m,
                         const float* __restrict__ v, int n) {
  int idx = blockIdx.x * blockDim.x + threadIdx.x;
  if (idx >= n) return;
  const int i = idx / HID;
  const int f = idx - i * HID;
  const float sl = dinv[i];
  float y = agg[idx] + h[idx] * sl * sl + b[f];
  y = fmaxf(y, 0.f);
  agg[idx] = g[f] * (y - m[f]) * rsqrtf(v[f] + BN_EPS) + be[f];
}

// ---------------------------------------------------------------- launcher
extern "C" void kernel_launch(void* const* d_in, const int* in_sizes, int n_in,
                              void* d_out, int out_size, void* d_ws, size_t ws_size,
                              hipStream_t stream) {
  const float* x   = (const float*)d_in[0];
  const int*   ei  = (const int*)d_in[1];
  const int*   src = ei;
  const int*   dst = ei + N_EDGES;

  const float *W1 = (const float*)d_in[2],  *b1 = (const float*)d_in[3];
  const float *g1 = (const float*)d_in[4],  *be1 = (const float*)d_in[5];
  const float *m1 = (const float*)d_in[6],  *v1 = (const float*)d_in[7];
  const float *W2 = (const float*)d_in[8],  *b2 = (const float*)d_in[9];
  const float *g2 = (const float*)d_in[10], *be2 = (const float*)d_in[11];
  const float *m2 = (const float*)d_in[12], *v2 = (const float*)d_in[13];
  const float *W3 = (const float*)d_in[14], *b3 = (const float*)d_in[15];
  const float *g3 = (const float*)d_in[16], *be3 = (const float*)d_in[17];
  const float *m3 = (const float*)d_in[18], *v3 = (const float*)d_in[19];
  const float *fW1 = (const float*)d_in[20], *fb1 = (const float*)d_in[21];
  const float *fg1 = (const float*)d_in[22], *fbe1 = (const float*)d_in[23];
  const float *fm1 = (const float*)d_in[24], *fv1 = (const float*)d_in[25];
  const float *fW2 = (const float*)d_in[26], *fb2 = (const float*)d_in[27];
  const float *fg2 = (const float*)d_in[28], *fbe2 = (const float*)d_in[29];
  const float *fm2 = (const float*)d_in[30], *fv2 = (const float*)d_in[31];
  const float *fW3 = (const float*)d_in[32], *fb3 = (const float*)d_in[33];
  float* out = (float*)d_out;

  // workspace: BUF0 (h), BUF1 (agg/activation), dinv
  float* buf0 = (float*)d_ws;
  float* buf1 = buf0 + (size_t)N_NODES * HID;
  float* dinv = buf1 + (size_t)N_NODES * HID;

  const int NH       = N_NODES * HID;
  const int mBlocks  = N_NODES / 16;                // 6250, exact
  const int edBlocks = (N_EDGES + 7) / 8;           // 1 wave per edge, 8 waves/block

  // degrees -> dinv
  fill_f32<<<(N_NODES + 255) / 256, 256, 0, stream>>>(dinv, N_NODES, 1.0f);
  degree_kernel<<<(N_EDGES + 255) / 256, 256, 0, stream>>>(dst, N_EDGES, dinv);
  rsqrt_kernel<<<(N_NODES + 255) / 256, 256, 0, stream>>>(dinv, N_NODES);

  // ---- GCN layer 1 (input x, K=75)
  gemm_wmma<F_INF, HID, 0><<<mBlocks, 256, 0, stream>>>(x, W1, buf0,
      nullptr, nullptr, nullptr, nullptr, nullptr);
  fill_f32<<<(NH + 255) / 256, 256, 0, stream>>>(buf1, NH, 0.0f);
  edge_agg<<<edBlocks, 256, 0, stream>>>(src, dst, N_EDGES, dinv, buf0, buf1);
  gcn_post<<<(NH + 255) / 256, 256, 0, stream>>>(buf1, buf0, dinv, b1, g1, be1, m1, v1, NH);

  // ---- GCN layer 2
  gemm_wmma<HID, HID, 0><<<mBlocks, 256, 0, stream>>>(buf1, W2, buf0,
      nullptr, nullptr, nullptr, nullptr, nullptr);
  fill_f32<<<(NH + 255) / 256, 256, 0, stream>>>(buf1, NH, 0.0f);
  edge_agg<<<edBlocks, 256, 0, stream>>>(src, dst, N_EDGES, dinv, buf0, buf1);
  gcn_post<<<(NH + 255) / 256, 256, 0, stream>>>(buf1, buf0, dinv, b2, g2, be2, m2, v2, NH);

  // ---- GCN layer 3
  gemm_wmma<HID, HID, 0><<<mBlocks, 256, 0, stream>>>(buf1, W3, buf0,
      nullptr, nullptr, nullptr, nullptr, nullptr);
  fill_f32<<<(NH + 255) / 256, 256, 0, stream>>>(buf1, NH, 0.0f);
  edge_agg<<<edBlocks, 256, 0, stream>>>(src, dst, N_EDGES, dinv, buf0, buf1);
  gcn_post<<<(NH + 255) / 256, 256, 0, stream>>>(buf1, buf0, dinv, b3, g3, be3, m3, v3, NH);

  // ---- Dense head: relu(BN(h@W+b)) x2, then relu(h@W+b)
  gemm_wmma<HID, HID, 1><<<mBlocks, 256, 0, stream>>>(buf1, fW1, buf0,
      fb1, fg1, fbe1, fm1, fv1);
  gemm_wmma<HID, HID, 1><<<mBlocks, 256, 0, stream>>>(buf0, fW2, buf1,
      fb2, fg2, fbe2, fm2, fv2);
  gemm_wmma<HID, OUTF, 2><<<mBlocks, 256, 0, stream>>>(buf1, fW3, out,
      fb3, nullptr, nullptr, nullptr, nullptr);
}